// LIFNodeTriton_80539226734729
// MI455X (gfx1250) — compile-verified
//
#include <hip/hip_runtime.h>

// LIF scan: v = v*0.5 + x_t; s = (v>=1); v -= s.  Pure HBM stream, ~516MB,
// roofline ~22us @ 23.3TB/s.  One float4 column per thread, v held in VGPRs;
// time-slices prefetched 8-deep via GLOBAL_LOAD_ASYNC_TO_LDS_B128 with NT
// hint (ASYNCcnt-tracked, in-order completion), spikes written with
// non-temporal b128 stores.  Epilogue is peeled with literal, decreasing
// s_wait_asynccnt counts so every consume is ordered behind its producer.

typedef float v4f __attribute__((ext_vector_type(4)));
typedef int   v4i __attribute__((vector_size(16)));    // matches builtin param
typedef __attribute__((address_space(3))) v4i las_v4i; // LDS-side pointee

#define N_     16
#define T_     64
#define CHW    65536          // 256*16*16 floats per (n,t) slice
#define COLS4  (CHW / 4)      // 16384 float4 columns per slice
#define DECAY  0.5f
#define VTH    1.0f
#define BLOCK  256
#define NBUF   8              // async pipeline depth (8 * 4KB = 32KB LDS/block)
#define CPOL_NT 1             // TH[2:0]=1 (non-temporal), scope WGP

#if defined(__HIP_DEVICE_COMPILE__) && \
    __has_builtin(__builtin_amdgcn_global_load_async_to_lds_b128)
#define USE_ASYNC 1
#else
#define USE_ASYNC 0
#endif

#if defined(__HIP_DEVICE_COMPILE__) && __has_builtin(__builtin_amdgcn_s_wait_asynccnt)
#define WAIT_ASYNC(n) __builtin_amdgcn_s_wait_asynccnt(n)
#else
#define WAIT_ASYNC(n) asm volatile("s_wait_asynccnt %0" ::"i"(n) : "memory")
#endif

__device__ __forceinline__ v4f lif_step(v4f& v, v4f xv) {
    v = v * DECAY + xv;
    v4f s;
    s.x = (v.x >= VTH) ? 1.0f : 0.0f;
    s.y = (v.y >= VTH) ? 1.0f : 0.0f;
    s.z = (v.z >= VTH) ? 1.0f : 0.0f;
    s.w = (v.w >= VTH) ? 1.0f : 0.0f;
    v -= s;   // reset by subtraction (VTH == 1)
    return s;
}

__global__ __launch_bounds__(BLOCK) void lif_scan_kernel(
    const float* __restrict__ x, const float* __restrict__ vinit,
    float* __restrict__ out) {
    const int g   = blockIdx.x * BLOCK + threadIdx.x;  // column id [0, 262144)
    const int n   = g >> 14;                           // g / COLS4
    const int cid = g & (COLS4 - 1);

    const v4f* __restrict__ x4 = (const v4f*)x;
    v4f* __restrict__       o4 = (v4f*)out;
    const size_t base = (size_t)n * T_ * COLS4 + (size_t)cid;

    v4f v = ((const v4f*)vinit)[g];

#if USE_ASYNC
    __shared__ v4f tile[NBUF][BLOCK];

    // Prologue: fill the 8-deep LDS ring with t = 0..7 slices.
#pragma unroll
    for (int t = 0; t < NBUF; ++t) {
        __builtin_amdgcn_global_load_async_to_lds_b128(
            (v4i*)(x4 + base + (size_t)t * COLS4),
            (las_v4i*)&tile[t][threadIdx.x], 0, CPOL_NT);
    }

    // Steady state: exactly NBUF loads outstanding at each wait, so
    // asynccnt <= NBUF-1 proves the oldest slice has landed (in-order).
#pragma unroll 1
    for (int t = 0; t < T_ - NBUF; ++t) {
        const int b = t & (NBUF - 1);
        WAIT_ASYNC(NBUF - 1);
        asm volatile("" ::: "memory");
        v4f xv = tile[b][threadIdx.x];  // each lane reads back its own 16B
        v4f s  = lif_step(v, xv);
        __builtin_nontemporal_store(s, o4 + base + (size_t)t * COLS4);
        asm volatile("" ::: "memory");  // keep ds_load before refill issue
        __builtin_amdgcn_global_load_async_to_lds_b128(
            (v4i*)(x4 + base + (size_t)(t + NBUF) * COLS4),
            (las_v4i*)&tile[b][threadIdx.x], 0, CPOL_NT);
    }

    // Epilogue: no refills, so outstanding count shrinks; wait counts must
    // shrink with it (immediates required by s_wait_asynccnt).
#define EPI(j)                                                               \
    {                                                                        \
        WAIT_ASYNC(NBUF - 1 - (j));                                          \
        asm volatile("" ::: "memory");                                       \
        const int t = T_ - NBUF + (j);                                       \
        v4f xv = tile[t & (NBUF - 1)][threadIdx.x];                          \
        v4f s  = lif_step(v, xv);                                            \
        __builtin_nontemporal_store(s, o4 + base + (size_t)t * COLS4);       \
    }
    EPI(0) EPI(1) EPI(2) EPI(3) EPI(4) EPI(5) EPI(6) EPI(7)
#undef EPI
#else
    // Fallback: 8-wide software pipeline of direct NT b128 loads.
#pragma unroll 1
    for (int to = 0; to < T_; to += NBUF) {
        v4f xs[NBUF];
#pragma unroll
        for (int j = 0; j < NBUF; ++j)
            xs[j] = __builtin_nontemporal_load(x4 + base + (size_t)(to + j) * COLS4);
#pragma unroll
        for (int j = 0; j < NBUF; ++j) {
            v4f s = lif_step(v, xs[j]);
            __builtin_nontemporal_store(s, o4 + base + (size_t)(to + j) * COLS4);
        }
    }
#endif
}

extern "C" void kernel_launch(void* const* d_in, const int* in_sizes, int n_in,
                              void* d_out, int out_size, void* d_ws, size_t ws_size,
                              hipStream_t stream) {
    (void)in_sizes; (void)n_in; (void)out_size; (void)d_ws; (void)ws_size;
    const float* x     = (const float*)d_in[0];
    const float* vinit = (const float*)d_in[1];
    float*       out   = (float*)d_out;

    const int total_cols = N_ * COLS4;          // 262144 float4 columns
    dim3 grid(total_cols / BLOCK);              // 1024 blocks
    lif_scan_kernel<<<grid, dim3(BLOCK), 0, stream>>>(x, vinit, out);
}